// ConvKAN_54082228191465
// MI455X (gfx1250) — compile-verified
//
#include <hip/hip_runtime.h>
#include <hip/hip_bf16.h>

typedef __attribute__((ext_vector_type(2))) float v2f;
typedef __attribute__((ext_vector_type(8))) float v8f;

#define HW      3136
#define WIDTH   56
#define CIN     64
#define COUT    128
#define P_TOT   25088                 // 8 * 56 * 56
#define KRED    6912                  // 64 * 9 * 12
#define W2_ELEMS (KRED * COUT)        // 884736

// ---------------------------------------------------------------------------
// Kernel 1: build B matrix W2[(c*9+k)*12+n][o]:
//   rows n=0..10 : control_points[o][c][k][n]   (spline weights)
//   row  n=11    : conv_w[o][c][k]              (conv folded into same GEMM)
// 3.5 MB -> stays resident in the 192MB L2 across all GEMM workgroups.
// ---------------------------------------------------------------------------
__global__ __launch_bounds__(256) void kan_prep_w2(
    const float* __restrict__ cp, const float* __restrict__ cw,
    float* __restrict__ W2)
{
    int idx = blockIdx.x * 256 + threadIdx.x;        // < 884736 exact
    int o = idx & 127;
    int r = idx >> 7;
    int n  = r % 12;
    int ck = r / 12;
    int k  = ck % 9;
    int c  = ck / 9;
    float v = (n < 11) ? cp[((o * CIN + c) * 9 + k) * 11 + n]
                       : cw[(o * CIN + c) * 9 + k];
    W2[idx] = v;
}

// Uniform-knot (j/11) degree-3 Cox-de Boor; 11 basis values + raw v (conv row).
__device__ __forceinline__ void kan_basis12(float v, float* __restrict__ dst)
{
    const float u = 1.0f / (1.0f + __expf(-v));
    const float w = 11.0f * u;
    float bs[11];
#pragma unroll
    for (int n = 0; n < 11; ++n)
        bs[n] = ((w >= (float)n) && (w < (float)(n + 1))) ? 1.0f : 0.0f;
#pragma unroll
    for (int d = 1; d <= 3; ++d) {
        const float inv = 1.0f / (float)d;
#pragma unroll
        for (int j = 0; j + d < 11; ++j)
            bs[j] = ((w - (float)j) * bs[j] +
                     ((float)(j + d + 1) - w) * bs[j + 1]) * inv;
    }
#pragma unroll
    for (int n = 0; n < 11; ++n) dst[n] = bs[n];
    dst[11] = v;
}

// ---------------------------------------------------------------------------
// Kernel 2: fused implicit GEMM  [25088 x 6912] x [6912 x 128] in fp32 WMMA.
// Workgroup: 64 pixels x 128 channels, 8 waves, each wave a 16x64 sub-tile.
// Ping-pong A buffers: basis VALU + x loads for iter i+1 overlap the WMMA
// stream of iter i (XDL WMMA co-executes with VALU); one barrier per iter.
// ---------------------------------------------------------------------------
__global__ __launch_bounds__(256) void kan_gemm(
    const float* __restrict__ x, const float* __restrict__ W2,
    float* __restrict__ tmp)
{
    __shared__ float A0[64][100];                // [pixel][96 rows + pad]
    __shared__ float A1[64][100];

    const int tid   = threadIdx.x;
    const int lane  = tid & 31;
    const int wave  = tid >> 5;
    const int msub  = wave & 3;                  // M sub-tile (16 pixels)
    const int obase = (wave >> 2) * 64;          // N range (64 channels)
    const int half  = lane >> 4;                 // half-wave select
    const int ln    = lane & 15;

    // fill-phase coordinates: thread -> (pixel, 2 channel slots)
    const int p_loc = tid & 63;
    const int cc0   = tid >> 6;                  // handles cc0 and cc0+4
    const int p_glb = blockIdx.x * 64 + p_loc;
    const int bimg  = p_glb / HW;
    const int rem   = p_glb - bimg * HW;
    const int py    = rem / WIDTH;
    const int px    = rem - py * WIDTH;

    v8f acc0 = {}, acc1 = {}, acc2 = {}, acc3 = {};

    // x gather for iteration it = k*8+cg, channel slot s (pad=0 pre-sigmoid)
    auto xload = [&](int it, int s) -> float {
        const int k  = it >> 3;
        const int cg = it & 7;
        const int yy = py + (k / 3) - 1;
        const int xx = px + (k % 3) - 1;
        const bool inb = (yy >= 0) & (yy < WIDTH) & (xx >= 0) & (xx < WIDTH);
        const int c  = cg * 8 + cc0 + s * 4;
        return inb ? x[((size_t)(bimg * CIN + c) * WIDTH + yy) * WIDTH + xx]
                   : 0.0f;
    };

    // ---- prologue: fill buffer 0 with iteration 0
    {
        float v0 = xload(0, 0);
        float v1 = xload(0, 1);
        kan_basis12(v0, &A0[p_loc][cc0 * 12]);
        kan_basis12(v1, &A0[p_loc][(cc0 + 4) * 12]);
    }
    __syncthreads();

    for (int it = 0; it < 72; ++it) {
        const int k  = it >> 3;
        const int cg = it & 7;
        float (*cur)[100] = (it & 1) ? A1 : A0;
        float (*nxt)[100] = (it & 1) ? A0 : A1;

        // issue next iteration's global x loads early (latency hides under WMMA)
        float n0 = 0.0f, n1 = 0.0f;
        if (it < 71) { n0 = xload(it + 1, 0); n1 = xload(it + 1, 1); }

        // ---- consume current buffer: 24 K-chunks of 4, 4 N-sub-tiles each
        // A frag: lane ln reads rows (2*half, 2*half+1) at pixel msub*16+ln
        // (ISA 32-bit A 16x4 layout: VGPR0 K=0|2, VGPR1 K=1|3); aligned b64.
        const float* arow = &cur[msub * 16 + ln][half * 2];
        // W2 row = ((cg*8 + kk/3)*9 + k)*12 + (kk%3)*4 (+2*half via lanes)
        const float* bbase =
            W2 + cg * 110592 + k * 1536 + obase + ln + half * 256;
#pragma unroll
        for (int kk = 0; kk < 24; ++kk) {
            v2f a = *(const v2f*)(arow + kk * 4);
            const float* bp = bbase + (kk / 3) * 13824 + (kk % 3) * 512;
            v2f b0 = { bp[0],  bp[128] };   // B 4x16: VGPR0 rows 0|2, VGPR1 rows 1|3
            v2f b1 = { bp[16], bp[144] };
            v2f b2 = { bp[32], bp[160] };
            v2f b3 = { bp[48], bp[176] };
            acc0 = __builtin_amdgcn_wmma_f32_16x16x4_f32(false, a, false, b0, (short)0, acc0, false, false);
            acc1 = __builtin_amdgcn_wmma_f32_16x16x4_f32(false, a, false, b1, (short)0, acc1, false, false);
            acc2 = __builtin_amdgcn_wmma_f32_16x16x4_f32(false, a, false, b2, (short)0, acc2, false, false);
            acc3 = __builtin_amdgcn_wmma_f32_16x16x4_f32(false, a, false, b3, (short)0, acc3, false, false);
        }

        // ---- fill the other buffer for iteration it+1 (no barrier needed
        // before: disjoint buffer; co-executes with the WMMA stream above)
        if (it < 71) {
            kan_basis12(n0, &nxt[p_loc][cc0 * 12]);
            kan_basis12(n1, &nxt[p_loc][(cc0 + 4) * 12]);
        }
        __syncthreads();
    }

    // ---- write pre-BN tile to tmp[pixel][channel]
    // C/D layout: VGPR j -> M = j (lanes 0-15) / j+8 (lanes 16-31), N = ln.
    const int row0 = blockIdx.x * 64 + msub * 16 + half * 8;
#pragma unroll
    for (int j = 0; j < 8; ++j) {
        float* d = tmp + (size_t)(row0 + j) * COUT + obase + ln;
        d[0]  = acc0[j];
        d[16] = acc1[j];
        d[32] = acc2[j];
        d[48] = acc3[j];
    }
}

// ---------------------------------------------------------------------------
// Kernel 3: per-channel mean / biased variance over 25088 pixels.
// (conv_b cancels exactly under batch-norm's mean subtraction -> omitted.)
// ---------------------------------------------------------------------------
__global__ __launch_bounds__(256) void kan_stats(
    const float* __restrict__ tmp, float* __restrict__ stats)
{
    __shared__ float s1[256];
    __shared__ float s2[256];
    const int o = blockIdx.x;
    const int t = threadIdx.x;
    float s = 0.0f, ss = 0.0f;
    for (int p = t; p < P_TOT; p += 256) {
        float v = tmp[(size_t)p * COUT + o];
        s += v; ss += v * v;
    }
    s1[t] = s; s2[t] = ss;
    __syncthreads();
    for (int d = 128; d > 0; d >>= 1) {
        if (t < d) { s1[t] += s1[t + d]; s2[t] += s2[t + d]; }
        __syncthreads();
    }
    if (t == 0) {
        float mean = s1[0] * (1.0f / (float)P_TOT);
        float var  = s2[0] * (1.0f / (float)P_TOT) - mean * mean;
        stats[o] = mean;
        stats[COUT + o] = var;
    }
}

// ---------------------------------------------------------------------------
// Kernel 4: normalize + affine, write NCHW output.
// ---------------------------------------------------------------------------
__global__ __launch_bounds__(256) void kan_norm(
    const float* __restrict__ tmp, const float* __restrict__ stats,
    const float* __restrict__ gamma, const float* __restrict__ beta,
    float* __restrict__ out)
{
    const int o    = blockIdx.x & 127;
    const int bimg = blockIdx.x >> 7;
    const float mean = stats[o];
    const float var  = stats[COUT + o];
    const float sc = rsqrtf(var + 1e-5f) * gamma[o];
    const float sh = beta[o] - mean * sc;
    for (int i = threadIdx.x; i < HW; i += 256) {
        float v = tmp[((size_t)bimg * HW + i) * COUT + o];
        out[((size_t)bimg * COUT + o) * HW + i] = v * sc + sh;
    }
}

// ---------------------------------------------------------------------------
extern "C" void kernel_launch(void* const* d_in, const int* in_sizes, int n_in,
                              void* d_out, int out_size, void* d_ws, size_t ws_size,
                              hipStream_t stream) {
    const float* x     = (const float*)d_in[0];
    const float* cp    = (const float*)d_in[1];
    const float* cw    = (const float*)d_in[2];
    // d_in[3] = conv_b: a per-channel pre-BN shift cancels exactly in BN.
    const float* gamma = (const float*)d_in[4];
    const float* beta  = (const float*)d_in[5];

    float* W2    = (float*)d_ws;
    float* tmp   = (float*)((char*)d_ws + (size_t)W2_ELEMS * 4);
    float* stats = (float*)((char*)d_ws + (size_t)W2_ELEMS * 4
                                        + (size_t)P_TOT * COUT * 4);

    kan_prep_w2<<<W2_ELEMS / 256, 256, 0, stream>>>(cp, cw, W2);
    kan_gemm<<<P_TOT / 64, 256, 0, stream>>>(x, W2, tmp);
    kan_stats<<<COUT, 256, 0, stream>>>(tmp, stats);
    kan_norm<<<8 * COUT, 256, 0, stream>>>(tmp, stats, gamma, beta, (float*)d_out);
}